// QueryGenerator_46265387712577
// MI455X (gfx1250) — compile-verified
//
#include <hip/hip_runtime.h>
#include <math.h>

// ---------------------------------------------------------------------------
// GRU LM forward for MI455X (gfx1250, wave32, WMMA).
//   V=128, E=100, H=1024, B=64, T=512
// Strategy:
//   1) embW = emb @ Wx + b_in   (128 x 3072)  -- tiny, replaces the x@Wx GEMM
//   2) 512x gru_step: rec = h @ Uh via V_WMMA_F32_16X16X4_F32 (exact f32).
//      Per step: grid=64 N-tiles, block=24 waves = (4 Mtiles x 3 gates x
//      2 K-halves) -> 1536 waves/step for latency hiding. Partial gate tiles
//      meet in LDS; epilogue waves fuse sigmoid/relu and write seq[:,t,:].
//   3) logits = seq @ Wd + bd via V_WMMA_F32_16X16X4_F32.
// ---------------------------------------------------------------------------

typedef __attribute__((ext_vector_type(2))) float v2f;
typedef __attribute__((ext_vector_type(8))) float v8f;

#define NV   128      // vocab
#define NE   100      // embed dim
#define NH   1024     // hidden
#define NH3  3072     // 3*H
#define NB   64       // batch
#define NT   512      // seq len

__device__ __forceinline__ v8f wmma_f32(v2f a, v2f b, v8f c) {
  // D = A(16x4, f32) x B(4x16, f32) + C(16x16, f32)
  return __builtin_amdgcn_wmma_f32_16x16x4_f32(
      false, a, false, b, (short)0, c, false, false);
}

__device__ __forceinline__ float sigmoidf_(float x) {
  return 1.0f / (1.0f + __expf(-x));
}

// ---------------------------------------------------------------------------
// Kernel 1: embW[v][j] = sum_e emb[v][e] * Wx[e][j] + b_in[j]
// ---------------------------------------------------------------------------
__global__ __launch_bounds__(256) void embw_kernel(
    const float* __restrict__ emb, const float* __restrict__ Wx,
    const float* __restrict__ b_in, float* __restrict__ embW) {
  int idx = blockIdx.x * 256 + threadIdx.x;          // [0, 128*3072)
  int v = idx / NH3;
  int j = idx - v * NH3;
  float acc = b_in[j];
  const float* er = emb + v * NE;
#pragma unroll 4
  for (int e = 0; e < NE; ++e)
    acc = fmaf(er[e], Wx[(size_t)e * NH3 + j], acc);
  embW[idx] = acc;
}

// ---------------------------------------------------------------------------
// Kernel 2: one GRU timestep.
//   grid  = 64 (16-wide H column tiles)
//   block = 768 threads = 24 waves: wave w -> kh = w/12 (K half),
//           r = w%12, gate g = r%3, Mtile m = r/3.
//   Each wave: 128 K-steps of V_WMMA_F32_16X16X4_F32 into one gate tile.
//   LDS: red[2][12][32] v8f partial tiles; epilogue waves 0..3 fuse gates.
// ---------------------------------------------------------------------------
__global__ __launch_bounds__(768) void gru_step_kernel(
    const float* __restrict__ Uh, const float* __restrict__ b_rec,
    const float* __restrict__ embW, const int* __restrict__ tokens,
    float* __restrict__ seq, int t) {
  __shared__ v8f red[2][12][32];   // [K-half][Mtile*3+gate][lane] = 24.5 KB

  const int lane  = threadIdx.x & 31;
  const int w     = threadIdx.x >> 5;        // 0..23
  const int kh    = w / 12;                  // K half: 0 or 1
  const int r12   = w - kh * 12;             // 0..11
  const int g     = r12 % 3;                 // gate: 0=z 1=r 2=h
  const int m     = r12 / 3;                 // Mtile: 0..3
  const int n0    = blockIdx.x * 16;         // H column tile base
  const int b0    = m * 16;                  // batch row base
  const int mrow  = lane & 15;               // A-frag row within tile
  const int ncol  = lane & 15;               // B/C-frag column within tile
  const int khalf = (lane >> 4) * 2;         // lanes 16..31 hold K+2,K+3

  v8f acc = {};

  if (t > 0) {  // wave-uniform: EXEC stays all-ones around WMMA
    const float* Arow =
        seq + ((size_t)(b0 + mrow) * NT + (t - 1)) * NH + kh * (NH / 2);
    const float* Bcol =
        Uh + (size_t)(kh * (NH / 2)) * NH3 + (size_t)g * NH + n0 + ncol;
    for (int k = 0; k < NH / 2; k += 4) {
      v2f a;
      a.x = Arow[k + khalf];
      a.y = Arow[k + khalf + 1];
      const float* bp = Bcol + (size_t)(k + khalf) * NH3;
      __builtin_prefetch(bp + (size_t)8 * NH3, 0, 1);   // global_prefetch_b8
      v2f b;
      b.x = bp[0];
      b.y = bp[NH3];
      acc = wmma_f32(a, b, acc);
    }
  }
  // (t == 0: h_prev = 0 -> rec = 0; accumulator stays zero.)

  red[kh][r12][lane] = acc;        // two ds_store_b128 per lane
  __syncthreads();

  // Epilogue: waves 0..3 (one per Mtile) fuse the three gates.
  if (w < 4) {
    const int em = w;              // Mtile handled by this wave
    v8f az = red[0][em * 3 + 0][lane] + red[1][em * 3 + 0][lane];
    v8f ar = red[0][em * 3 + 1][lane] + red[1][em * 3 + 1][lane];
    v8f ah = red[0][em * 3 + 2][lane] + red[1][em * 3 + 2][lane];

    const int n   = n0 + ncol;
    const float brz = b_rec[n];
    const float brr = b_rec[NH + n];
    const float brh = b_rec[2 * NH + n];
    const int eb0 = em * 16;
    const int mhi = (lane >> 4) << 3;

#pragma unroll
    for (int i = 0; i < 8; ++i) {
      const int b   = eb0 + i + mhi;
      const int tok = tokens[b * NT + t];
      const float* ew = embW + (size_t)tok * NH3;
      const float z  = sigmoidf_(ew[n]      + az[i] + brz);
      const float rg = sigmoidf_(ew[NH + n] + ar[i] + brr);
      const float hh = fmaxf(0.0f, fmaf(rg, ah[i] + brh, ew[2 * NH + n]));
      const float hp = (t == 0) ? 0.0f
                                : seq[((size_t)b * NT + (t - 1)) * NH + n];
      const float hn = fmaf(z, hp - hh, hh);   // z*hp + (1-z)*hh
      seq[((size_t)b * NT + t) * NH + n] = hn;
    }
  }
}

// ---------------------------------------------------------------------------
// Kernel 3: logits = seq @ Wd + bd.  M = B*T = 32768, N = 128, K = 1024.
// grid = 512 blocks (64 rows each), block = 128 threads = 4 waves.
// Wave m: row tile m; one A fragment feeds all 8 N-tiles (full vocab).
// ---------------------------------------------------------------------------
__global__ __launch_bounds__(128) void logits_kernel(
    const float* __restrict__ seq, const float* __restrict__ Wd,
    const float* __restrict__ bd, float* __restrict__ out) {
  const int lane  = threadIdx.x & 31;
  const int wave  = threadIdx.x >> 5;
  const size_t r0 = (size_t)blockIdx.x * 64 + (size_t)wave * 16;
  const int mrow  = lane & 15;
  const int ncol  = lane & 15;
  const int khalf = (lane >> 4) * 2;

  v8f acc[8] = {};  // 8 N-tiles of 16 cols = full V=128

  const float* Arow = seq + (r0 + mrow) * NH;
  for (int k = 0; k < NH; k += 4) {
    v2f a;
    a.x = Arow[k + khalf];
    a.y = Arow[k + khalf + 1];
    const float* bp = Wd + (size_t)(k + khalf) * NV + ncol;
#pragma unroll
    for (int nt = 0; nt < 8; ++nt) {
      v2f b;
      b.x = bp[nt * 16];
      b.y = bp[NV + nt * 16];
      acc[nt] = wmma_f32(a, b, acc[nt]);
    }
  }

  const int mhi = (lane >> 4) << 3;
#pragma unroll
  for (int nt = 0; nt < 8; ++nt) {
    const int n = nt * 16 + ncol;
    const float bias = bd[n];
#pragma unroll
    for (int i = 0; i < 8; ++i) {
      const size_t row = r0 + (size_t)(i + mhi);
      out[row * NV + n] = acc[nt][i] + bias;
    }
  }
}

// ---------------------------------------------------------------------------
// Host launcher. Inputs (setup_inputs order):
//   0 tokens(i32 B*T) 1 emb 2 Wx 3 Uh 4 b_in 5 b_rec 6 Wd 7 bd
// Workspace: embW (128*3072 f32 = 1.57MB) + seq (64*512*1024 f32 = 134MB).
// ---------------------------------------------------------------------------
extern "C" void kernel_launch(void* const* d_in, const int* in_sizes, int n_in,
                              void* d_out, int out_size, void* d_ws,
                              size_t ws_size, hipStream_t stream) {
  const int*   tokens = (const int*)d_in[0];
  const float* emb    = (const float*)d_in[1];
  const float* Wx     = (const float*)d_in[2];
  const float* Uh     = (const float*)d_in[3];
  const float* b_in   = (const float*)d_in[4];
  const float* b_rec  = (const float*)d_in[5];
  const float* Wd     = (const float*)d_in[6];
  const float* bd     = (const float*)d_in[7];
  float* out = (float*)d_out;

  float* embW = (float*)d_ws;                       // 128*3072 floats
  float* seq  = embW + (size_t)NV * NH3;            // 64*512*1024 floats

  // 1) fold emb @ Wx + b_in into a 128-row lookup table
  embw_kernel<<<(NV * NH3) / 256, 256, 0, stream>>>(emb, Wx, b_in, embW);

  // 2) sequential recurrence: one launch per timestep (graph-captured)
  for (int t = 0; t < NT; ++t)
    gru_step_kernel<<<NH / 16, 768, 0, stream>>>(Uh, b_rec, embW, tokens,
                                                 seq, t);

  // 3) decoder GEMM
  logits_kernel<<<(NB * NT) / 64, 128, 0, stream>>>(seq, Wd, bd, out);
}